// SVDKMuyGPs_Heaton_4818953307045
// MI455X (gfx1250) — compile-verified
//
#include <hip/hip_runtime.h>
#include <math.h>

typedef __attribute__((ext_vector_type(2))) float v2f;
typedef __attribute__((ext_vector_type(8))) float v8f;

#define NNB 64          // neighbors per batch point
#define EMB_D 10        // embedding dim
#define EMB_P 12        // padded to multiple of 4 for WMMA K-tiling
#define KEPS 1e-3f
#define SQRT3F 1.7320508075688772f
#define FPITCH 65       // 64x64 matrix with +1 pad to dodge LDS bank conflicts

// ---------------- Kernel 1: embedding MLP  emb[n] = prelu(W2 prelu(W1 x + b1) + b2)
__global__ __launch_bounds__(256) void embed_kernel(
    const float* __restrict__ x, const float* __restrict__ W1, const float* __restrict__ b1,
    const float* __restrict__ a1, const float* __restrict__ W2, const float* __restrict__ b2,
    const float* __restrict__ a2, float* __restrict__ emb, int n)
{
    int i = blockIdx.x * 256 + threadIdx.x;
    if (i >= n) return;
    float x0 = x[2*i], x1 = x[2*i+1];
    float al1 = a1[0], al2 = a2[0];
    float h[30];
    #pragma unroll
    for (int j = 0; j < 30; ++j) {
        float z = fmaf(W1[2*j], x0, fmaf(W1[2*j+1], x1, b1[j]));
        h[j] = z > 0.f ? z : al1 * z;
    }
    #pragma unroll
    for (int o = 0; o < EMB_D; ++o) {
        float z = b2[o];
        #pragma unroll
        for (int j = 0; j < 30; ++j) z = fmaf(W2[o*30+j], h[j], z);
        emb[i*EMB_P + o] = z > 0.f ? z : al2 * z;
    }
    emb[i*EMB_P + 10] = 0.f;
    emb[i*EMB_P + 11] = 0.f;
}

__device__ __forceinline__ float matern32(float d) {
    float s = SQRT3F * d;
    return (1.f + s) * __expf(-s);
}

// ---------------- Kernel 2: one workgroup (8 waves) per batch element
__global__ __launch_bounds__(256) void krige_kernel(
    const float* __restrict__ emb,
    const int*   __restrict__ bidx,
    const int*   __restrict__ nnidx,
    const float* __restrict__ targets,   // [B, NNB, 1]
    float* __restrict__ out,             // [2B+1]
    float* __restrict__ partial,         // [B] sigma_sq partials
    int B)
{
    __shared__ float Xn[NNB * EMB_P];      // gathered neighbor embeddings (padded)
    __shared__ float xb[EMB_P];
    __shared__ float Fm[NNB * FPITCH];     // Gram -> kernel matrix -> Cholesky factor
    __shared__ float sq[NNB];
    __shared__ float Kc[NNB];
    __shared__ float y0[NNB];              // targets RHS -> coeffs
    __shared__ float y1[NNB];              // Kcross RHS -> solved_cross
    __shared__ float r0[NNB], r1[NNB], r2[NNB];

    const int b    = blockIdx.x;
    const int tid  = threadIdx.x;
    const int lane = tid & 31;
    const int wave = tid >> 5;

    // ---- gather neighbor embeddings + query embedding into LDS
    for (int idx = tid; idx < NNB * EMB_P; idx += 256) {
        int row = idx / EMB_P;
        Xn[idx] = emb[(long)nnidx[b*NNB + row] * EMB_P + (idx % EMB_P)];
    }
    if (tid < EMB_P) xb[tid] = emb[(long)bidx[b] * EMB_P + tid];
    __syncthreads();

    // ---- Gram G = Xn * Xn^T via V_WMMA_F32_16X16X4_F32 (4x4 tile grid, K padded to 12)
    // A 16x4 layout: lanes 0-15 hold M=lane {K=0,1}; lanes 16-31 hold M=lane-16 {K=2,3}.
    // For X*X^T, the B fragment (4x16) of tile column tj has the same per-lane contents
    // as the A fragment of tile row tj, so both load identically.
    const int mrow = lane & 15;
    const int koff = (lane >= 16) ? 2 : 0;
    for (int t = wave; t < 16; t += 8) {      // every wave runs both iterations, EXEC all-1s
        int ti = t >> 2, tj = t & 3;
        v8f c = {};
        #pragma unroll
        for (int kt = 0; kt < 3; ++kt) {
            int kk = kt*4 + koff;
            v2f af, bf;
            af.x = Xn[(ti*16 + mrow)*EMB_P + kk];
            af.y = Xn[(ti*16 + mrow)*EMB_P + kk + 1];
            bf.x = Xn[(tj*16 + mrow)*EMB_P + kk];
            bf.y = Xn[(tj*16 + mrow)*EMB_P + kk + 1];
            c = __builtin_amdgcn_wmma_f32_16x16x4_f32(
                    false, af, false, bf, (short)0, c, false, false);
        }
        // D layout: VGPR v -> row v (lanes 0-15) / row v+8 (lanes 16-31), col = lane&15
        int col   = tj*16 + mrow;
        int rbase = ti*16 + ((lane >= 16) ? 8 : 0);
        #pragma unroll
        for (int v = 0; v < 8; ++v)
            Fm[(rbase + v)*FPITCH + col] = c[v];
    }
    __syncthreads();

    // ---- squared norms from Gram diagonal
    if (tid < NNB) sq[tid] = Fm[tid*FPITCH + tid];
    __syncthreads();

    // ---- transform Gram -> Matern kernel matrix + nugget, in place
    for (int idx = tid; idx < NNB*NNB; idx += 256) {
        int i = idx >> 6, j = idx & 63;
        float g  = Fm[i*FPITCH + j];
        float d2 = fmaxf(sq[i] + sq[j] - 2.f*g, 0.f);
        float d  = sqrtf(d2 + 1e-12f);
        float kv = matern32(d);
        if (i == j) kv += KEPS;
        Fm[i*FPITCH + j] = kv;
    }
    // ---- crosswise kernel vector + RHS setup (independent arrays, no barrier needed yet)
    if (tid < NNB) {
        float acc = 0.f;
        #pragma unroll
        for (int d = 0; d < EMB_D; ++d) {
            float diff = xb[d] - Xn[tid*EMB_P + d];
            acc = fmaf(diff, diff, acc);
        }
        float dc = sqrtf(fmaxf(acc, 0.f) + 1e-12f);
        float kc = matern32(dc);
        Kc[tid] = kc;
        y0[tid] = targets[b*NNB + tid];
        y1[tid] = kc;
    }
    __syncthreads();

    // ---- Cholesky (right-looking; keep full symmetric trailing block updated)
    for (int k = 0; k < NNB; ++k) {
        if (tid == 0) Fm[k*FPITCH + k] = sqrtf(Fm[k*FPITCH + k]);
        __syncthreads();
        float dinv = 1.f / Fm[k*FPITCH + k];
        for (int i = k+1+tid; i < NNB; i += 256) Fm[i*FPITCH + k] *= dinv;
        __syncthreads();
        int m = NNB - 1 - k;
        for (int idx = tid; idx < m*m; idx += 256) {
            int i = k+1 + idx / m;
            int j = k+1 + idx % m;
            Fm[i*FPITCH + j] = fmaf(-Fm[i*FPITCH + k], Fm[j*FPITCH + k], Fm[i*FPITCH + j]);
        }
        __syncthreads();
    }

    // ---- forward solve L z = rhs (both RHS simultaneously)
    for (int k = 0; k < NNB; ++k) {
        if (tid == 0) { float d = Fm[k*FPITCH + k]; y0[k] /= d; y1[k] /= d; }
        __syncthreads();
        float z0 = y0[k], z1 = y1[k];
        for (int i = k+1+tid; i < NNB; i += 256) {
            float l = Fm[i*FPITCH + k];
            y0[i] = fmaf(-l, z0, y0[i]);
            y1[i] = fmaf(-l, z1, y1[i]);
        }
        __syncthreads();
    }
    // ---- backward solve L^T x = z
    for (int k = NNB-1; k >= 0; --k) {
        if (tid == 0) { float d = Fm[k*FPITCH + k]; y0[k] /= d; y1[k] /= d; }
        __syncthreads();
        float z0 = y0[k], z1 = y1[k];
        for (int i = tid; i < k; i += 256) {
            float l = Fm[k*FPITCH + i];       // L^T[i][k] = L[k][i]
            y0[i] = fmaf(-l, z0, y0[i]);
            y1[i] = fmaf(-l, z1, y1[i]);
        }
        __syncthreads();
    }

    // ---- reductions: prediction, variance, sigma_sq partial
    if (tid < NNB) {
        float tgt = targets[b*NNB + tid];
        r0[tid] = Kc[tid] * y0[tid];
        r1[tid] = Kc[tid] * y1[tid];
        r2[tid] = tgt     * y0[tid];
    }
    __syncthreads();
    if (tid == 0) {
        float s0 = 0.f, s1 = 0.f, s2 = 0.f;
        for (int i = 0; i < NNB; ++i) { s0 += r0[i]; s1 += r1[i]; s2 += r2[i]; }
        out[b]       = s0;          // prediction (R == 1)
        out[B + b]   = 1.f - s1;    // variance
        partial[b]   = s2;          // sigma_sq contribution
    }
}

// ---------------- Kernel 3: reduce sigma_sq partials
__global__ __launch_bounds__(256) void sigma_kernel(
    const float* __restrict__ partial, float* __restrict__ out, int B)
{
    __shared__ float red[256];
    float s = 0.f;
    for (int i = threadIdx.x; i < B; i += 256) s += partial[i];
    red[threadIdx.x] = s;
    __syncthreads();
    for (int w = 128; w > 0; w >>= 1) {
        if (threadIdx.x < w) red[threadIdx.x] += red[threadIdx.x + w];
        __syncthreads();
    }
    if (threadIdx.x == 0) out[2*B] = red[0] / ((float)NNB * (float)B);
}

extern "C" void kernel_launch(void* const* d_in, const int* in_sizes, int n_in,
                              void* d_out, int out_size, void* d_ws, size_t ws_size,
                              hipStream_t stream) {
    const float* x       = (const float*)d_in[0];
    const float* W1      = (const float*)d_in[1];
    const float* b1      = (const float*)d_in[2];
    const float* a1      = (const float*)d_in[3];
    const float* W2      = (const float*)d_in[4];
    const float* b2      = (const float*)d_in[5];
    const float* a2      = (const float*)d_in[6];
    const int*   bidx    = (const int*)d_in[7];
    const int*   nnidx   = (const int*)d_in[8];
    const float* targets = (const float*)d_in[9];
    float* out = (float*)d_out;

    int N = in_sizes[0] / 2;          // 100000
    int B = in_sizes[7];              // 8192

    float* emb     = (float*)d_ws;            // N * EMB_P floats
    float* partial = emb + (size_t)N * EMB_P; // B floats

    embed_kernel<<<(N + 255) / 256, 256, 0, stream>>>(x, W1, b1, a1, W2, b2, a2, emb, N);
    krige_kernel<<<B, 256, 0, stream>>>(emb, bidx, nnidx, targets, out, partial, B);
    sigma_kernel<<<1, 256, 0, stream>>>(partial, out, B);
}